// GAT_78761110274485
// MI455X (gfx1250) — compile-verified
//
#include <hip/hip_runtime.h>
#include <math.h>

#define N_NODES 100000
#define N_EDGES 1600000
#define E2      (N_EDGES + N_NODES)   // with self loops
#define F_IN    166
#define NH      4                     // heads
#define DH      16                    // dim per head
#define HD      64                    // NH*DH
#define NEG_SLOPE 0.2f
#define KMAX    168                   // F_IN padded to multiple of 4

typedef __attribute__((ext_vector_type(2))) float v2f;
typedef __attribute__((ext_vector_type(8))) float v8f;

// ---- hardware float atomics -------------------------------------------------
__device__ __forceinline__ void atomic_add_f32(float* p, float v) {
    // GLOBAL_ATOMIC_ADD_F32 (no return) -> STOREcnt; L2 atomic unit
    asm volatile("global_atomic_add_f32 %0, %1, off" :: "v"(p), "v"(v) : "memory");
}

__device__ __forceinline__ void atomic_max_f32(float* p, float v) {
    // sign-split monotonic-bits trick; correct for mixed signs, init = -inf
    if (v >= 0.0f) atomicMax((int*)p, __float_as_int(v));
    else           atomicMin((unsigned int*)p, (unsigned int)__float_as_int(v));
}

// ---- GEMM: Out[M][64] = A[M][K] @ W[K][64], via V_WMMA_F32_16X16X4_F32 ------
// LDS holds W with K-pairs interleaved: ldsW[(k>>1)*128 + n*2 + (k&1)]
// so each lane's B fragment {W[kA][n], W[kA+1][n]} is one ds_load_b64.
__global__ __launch_bounds__(256)
void gemm_wmma_kernel(const float* __restrict__ A, const float* __restrict__ W,
                      float* __restrict__ Out, int M, int K) {
    __shared__ float ldsW[KMAX * HD];               // 43008 B max
    const int Kpad = (K + 3) & ~3;
    for (int idx = threadIdx.x; idx < Kpad * HD; idx += 256) {
        int k = idx / HD, n = idx % HD;
        float v = (k < K) ? W[k * HD + n] : 0.0f;
        ldsW[(k >> 1) * 128 + n * 2 + (k & 1)] = v;
    }
    __syncthreads();

    const int wave = threadIdx.x >> 5;
    const int lane = threadIdx.x & 31;
    const int half = lane >> 4;                      // K half select (A: K={0,1} vs {2,3})
    const int l16  = lane & 15;
    const int row0 = blockIdx.x * 128 + wave * 16;
    if (row0 >= M) return;                           // wave-uniform exit, EXEC stays full

    int rowA = row0 + l16; if (rowA >= M) rowA = M - 1;   // clamp (stores guarded)
    const float* arow = A + (long)rowA * K;

    v8f acc[4] = {};                                 // 4 column tiles = 4 heads
    const int Kfull = K & ~3;                        // unguarded main steps
    for (int k0 = 0; k0 < Kfull; k0 += 4) {
        const float2 av = *(const float2*)(arow + k0 + half * 2);   // 8B, aligned (K even)
        v2f a = {av.x, av.y};
        const float* wrow = &ldsW[((k0 >> 1) + half) * 128];
#pragma unroll
        for (int c = 0; c < 4; ++c) {
            const float2 bv = *(const float2*)(wrow + c * 32 + l16 * 2);
            v2f b = {bv.x, bv.y};
            acc[c] = __builtin_amdgcn_wmma_f32_16x16x4_f32(
                false, a, false, b, (short)0, acc[c], false, false);
        }
    }
    if (Kfull < K) {                                 // tail (K=166 -> 2 leftover cols)
        const int kA = Kfull + half * 2;
        v2f a;
        a.x = (kA     < K) ? arow[kA]     : 0.0f;
        a.y = (kA + 1 < K) ? arow[kA + 1] : 0.0f;
        const float* wrow = &ldsW[((Kfull >> 1) + half) * 128];  // zero-padded rows
#pragma unroll
        for (int c = 0; c < 4; ++c) {
            const float2 bv = *(const float2*)(wrow + c * 32 + l16 * 2);
            v2f b = {bv.x, bv.y};
            acc[c] = __builtin_amdgcn_wmma_f32_16x16x4_f32(
                false, a, false, b, (short)0, acc[c], false, false);
        }
    }
#pragma unroll
    for (int c = 0; c < 4; ++c)
#pragma unroll
        for (int r = 0; r < 8; ++r) {
            int row = row0 + r + 8 * half;
            if (row < M) Out[(long)row * HD + c * 16 + l16] = acc[c][r];
        }
}

// ---- per-(node,head) attention logits: al = <h[n,hh,:], a[hh,:]> ------------
__global__ void logits_kernel(const float* __restrict__ h,
                              const float* __restrict__ a_src,
                              const float* __restrict__ a_dst,
                              float* __restrict__ al_src,
                              float* __restrict__ al_dst) {
    int t = blockIdx.x * blockDim.x + threadIdx.x;
    if (t >= N_NODES * NH) return;
    int n = t >> 2, hh = t & 3;
    const float* hp = h + (long)n * HD + hh * DH;
    const float* as = a_src + hh * DH;
    const float* ad = a_dst + hh * DH;
    float s = 0.f, d = 0.f;
#pragma unroll
    for (int k = 0; k < DH; ++k) { float v = hp[k]; s += v * as[k]; d += v * ad[k]; }
    al_src[t] = s; al_dst[t] = d;
}

// ---- init: m = -inf, denom = 0, aggr = 0 ------------------------------------
__global__ void init_kernel(float* __restrict__ m, float* __restrict__ denom,
                            float* __restrict__ aggr) {
    int t = blockIdx.x * blockDim.x + threadIdx.x;
    if (t < N_NODES * HD) aggr[t] = 0.0f;
    if (t < N_NODES * NH) { m[t] = -3.402823466e38f; denom[t] = 0.0f; }
}

// ---- edge pass 1: segment max of LeakyReLU logits over dst ------------------
__global__ void edge_max_kernel(const int* __restrict__ ei,
                                const float* __restrict__ al_src,
                                const float* __restrict__ al_dst,
                                float* __restrict__ m) {
    int e = blockIdx.x * blockDim.x + threadIdx.x;
    if (e >= E2) return;
    int src, dst;
    if (e < N_EDGES) { src = ei[e]; dst = ei[N_EDGES + e]; }
    else             { src = dst = e - N_EDGES; }          // self loop
    const float4 as = *(const float4*)(al_src + src * 4);
    const float4 ad = *(const float4*)(al_dst + dst * 4);
    float v;
    v = as.x + ad.x; v = v > 0.f ? v : NEG_SLOPE * v; atomic_max_f32(&m[dst * 4 + 0], v);
    v = as.y + ad.y; v = v > 0.f ? v : NEG_SLOPE * v; atomic_max_f32(&m[dst * 4 + 1], v);
    v = as.z + ad.z; v = v > 0.f ? v : NEG_SLOPE * v; atomic_max_f32(&m[dst * 4 + 2], v);
    v = as.w + ad.w; v = v > 0.f ? v : NEG_SLOPE * v; atomic_max_f32(&m[dst * 4 + 3], v);
}

// ---- edge pass 2 (fused): w = exp(e - m[dst]); denom += w; aggr += h[src]*w -
__global__ void edge_accum_kernel(const int* __restrict__ ei,
                                  const float* __restrict__ al_src,
                                  const float* __restrict__ al_dst,
                                  const float* __restrict__ m,
                                  const float* __restrict__ h,
                                  float* __restrict__ denom,
                                  float* __restrict__ aggr) {
    int t = blockIdx.x * blockDim.x + threadIdx.x;
    if (t >= E2 * NH) return;
    int e = t >> 2, hh = t & 3;
    int src, dst;
    if (e < N_EDGES) { src = ei[e]; dst = ei[N_EDGES + e]; }
    else             { src = dst = e - N_EDGES; }
    float v = al_src[src * 4 + hh] + al_dst[dst * 4 + hh];
    v = v > 0.f ? v : NEG_SLOPE * v;
    float w = __expf(v - m[dst * 4 + hh]);
    atomic_add_f32(&denom[dst * 4 + hh], w);
    const float* hs = h + (long)src * HD + hh * DH;
    float* ap = aggr + (long)dst * HD + hh * DH;
#pragma unroll
    for (int d = 0; d < DH; d += 4) {
        float4 hv = *(const float4*)(hs + d);
        atomic_add_f32(ap + d + 0, hv.x * w);
        atomic_add_f32(ap + d + 1, hv.y * w);
        atomic_add_f32(ap + d + 2, hv.z * w);
        atomic_add_f32(ap + d + 3, hv.w * w);
    }
}

// ---- finalize: out = relu(aggr/denom + bias) (safe in-place) ----------------
__global__ void finalize_kernel(const float* __restrict__ aggr,
                                const float* __restrict__ denom,
                                const float* __restrict__ bias,
                                float* __restrict__ out) {
    int t = blockIdx.x * blockDim.x + threadIdx.x;
    if (t >= N_NODES * HD) return;
    int n = t >> 6, j = t & 63;
    float v = aggr[t] / denom[n * 4 + (j >> 4)] + bias[j];
    out[t] = v > 0.f ? v : 0.f;
}

// ---- MLP head: sigmoid((feat@lw1 + lb1)@lw2 + lb2) --------------------------
__global__ __launch_bounds__(256)
void mlp_kernel(const float* __restrict__ feat,
                const float* __restrict__ lw1, const float* __restrict__ lb1,
                const float* __restrict__ lw2, const float* __restrict__ lb2,
                float* __restrict__ out) {
    __shared__ float w1s[HD * DH];
    __shared__ float w2s[DH];
    __shared__ float b1s[DH];
    for (int i = threadIdx.x; i < HD * DH; i += 256) w1s[i] = lw1[i];
    if (threadIdx.x < DH) { w2s[threadIdx.x] = lw2[threadIdx.x]; b1s[threadIdx.x] = lb1[threadIdx.x]; }
    __syncthreads();
    int n = blockIdx.x * blockDim.x + threadIdx.x;
    if (n >= N_NODES) return;
    float acc[DH];
#pragma unroll
    for (int k = 0; k < DH; ++k) acc[k] = b1s[k];
    const float* f = feat + (long)n * HD;
    for (int j = 0; j < HD; ++j) {
        float xj = f[j];
#pragma unroll
        for (int k = 0; k < DH; ++k) acc[k] += xj * w1s[j * DH + k];
    }
    float o = lb2[0];
#pragma unroll
    for (int k = 0; k < DH; ++k) o += acc[k] * w2s[k];
    out[n] = 1.0f / (1.0f + __expf(-o));
}

extern "C" void kernel_launch(void* const* d_in, const int* in_sizes, int n_in,
                              void* d_out, int out_size, void* d_ws, size_t ws_size,
                              hipStream_t stream) {
    const float* x   = (const float*)d_in[0];
    const int*   ei  = (const int*)  d_in[1];
    const float* W1  = (const float*)d_in[2];
    const float* as1 = (const float*)d_in[3];
    const float* ad1 = (const float*)d_in[4];
    const float* b1  = (const float*)d_in[5];
    const float* W2  = (const float*)d_in[6];
    const float* as2 = (const float*)d_in[7];
    const float* ad2 = (const float*)d_in[8];
    const float* b2  = (const float*)d_in[9];
    const float* lw1 = (const float*)d_in[10];
    const float* lb1 = (const float*)d_in[11];
    const float* lw2 = (const float*)d_in[12];
    const float* lb2 = (const float*)d_in[13];
    float* out = (float*)d_out;
    (void)in_sizes; (void)n_in; (void)out_size; (void)ws_size;

    // workspace layout (floats). feat aliases aggr: finalize is elementwise
    // in-place, and the next GEMM consumes feat before init re-zeroes aggr.
    float* ws   = (float*)d_ws;
    float* hbuf = ws;                                   // N*HD
    float* aggr = hbuf + (size_t)N_NODES * HD;          // N*HD (also 'feat')
    float* feat = aggr;
    float* alS  = aggr + (size_t)N_NODES * HD;          // N*NH
    float* alD  = alS  + (size_t)N_NODES * NH;          // N*NH
    float* mbuf = alD  + (size_t)N_NODES * NH;          // N*NH
    float* dbuf = mbuf + (size_t)N_NODES * NH;          // N*NH

    dim3 blk(256);
    const int gGemm = (N_NODES + 127) / 128;            // 8 waves x 16 rows / block
    const int gLog  = (N_NODES * NH + 255) / 256;
    const int gInit = (N_NODES * HD + 255) / 256;
    const int gEm   = (E2 + 255) / 256;
    const int gEa   = (E2 * NH + 255) / 256;
    const int gMlp  = (N_NODES + 255) / 256;

    // ---- layer 1 ----
    gemm_wmma_kernel<<<gGemm, blk, 0, stream>>>(x, W1, hbuf, N_NODES, F_IN);
    logits_kernel   <<<gLog,  blk, 0, stream>>>(hbuf, as1, ad1, alS, alD);
    init_kernel     <<<gInit, blk, 0, stream>>>(mbuf, dbuf, aggr);
    edge_max_kernel <<<gEm,   blk, 0, stream>>>(ei, alS, alD, mbuf);
    edge_accum_kernel<<<gEa,  blk, 0, stream>>>(ei, alS, alD, mbuf, hbuf, dbuf, aggr);
    finalize_kernel <<<gInit, blk, 0, stream>>>(aggr, dbuf, b1, feat);

    // ---- layer 2 ----
    gemm_wmma_kernel<<<gGemm, blk, 0, stream>>>(feat, W2, hbuf, N_NODES, HD);
    logits_kernel   <<<gLog,  blk, 0, stream>>>(hbuf, as2, ad2, alS, alD);
    init_kernel     <<<gInit, blk, 0, stream>>>(mbuf, dbuf, aggr);
    edge_max_kernel <<<gEm,   blk, 0, stream>>>(ei, alS, alD, mbuf);
    edge_accum_kernel<<<gEa,  blk, 0, stream>>>(ei, alS, alD, mbuf, hbuf, dbuf, aggr);
    finalize_kernel <<<gInit, blk, 0, stream>>>(aggr, dbuf, b2, feat);

    // ---- head ----
    mlp_kernel<<<gMlp, blk, 0, stream>>>(feat, lw1, lb1, lw2, lb2, out);
}